// CompareGate_6700148982256
// MI455X (gfx1250) — compile-verified
//
#include <hip/hip_runtime.h>
#include <hip/hip_bf16.h>
#include <math.h>

typedef __attribute__((ext_vector_type(2))) float v2f;
typedef __attribute__((ext_vector_type(4))) float v4f;
typedef __attribute__((ext_vector_type(8))) float v8f;

#define B_   32
#define C_   512
#define HW   4096
#define KSEL 256
#define EPS  1e-8f

// ---------------- Pass 1: cosine-similarity stats via V_WMMA_F32_16X16X4_F32
// One workgroup per (batch, 16-channel block). 256 threads = 8 waves.
// Double-buffered LDS tiles filled by GLOBAL_LOAD_ASYNC_TO_LDS_B128 (ASYNCcnt),
// overlapped with the WMMA accumulation on the other buffer.
// Tile: 16 channels x TILE spatial floats, row stride RS = TILE+4 so that
// RS mod 64 == 4 -> the per-lane b64 reads for the WMMA A/B layout are
// bank-conflict free (lanes 0-15 at banks 4r..4r+1, lanes 16-31 at 4r+2..4r+3).
#define TILE 256
#define RS   260
#define NW   8
#define NT   (HW / TILE)          // 16 tiles
#define BUFF (2 * 16 * RS)        // floats per (xs|ys) double tile

__device__ __forceinline__ void async_fill_tile(
    const float* __restrict__ x, const float* __restrict__ y,
    float* xs, float* ys, size_t base, int tile, int t) {
    // 16 rows x 64 float4 per tensor, 256 threads -> 4 float4 each per tensor
    #pragma unroll
    for (int k = 0; k < 4; ++k) {
        int i  = t + k * 256;
        int r  = i >> 6;
        int c4 = i & 63;
        size_t g = base + (size_t)r * HW + (size_t)tile * TILE + (size_t)c4 * 4;
        unsigned long long gx = (unsigned long long)(uintptr_t)(x + g);
        unsigned long long gy = (unsigned long long)(uintptr_t)(y + g);
        unsigned int lx = (unsigned int)(uintptr_t)(xs + r * RS + c4 * 4);
        unsigned int ly = (unsigned int)(uintptr_t)(ys + r * RS + c4 * 4);
        asm volatile("global_load_async_to_lds_b128 %0, %1, off"
                     :: "v"(lx), "v"(gx) : "memory");
        asm volatile("global_load_async_to_lds_b128 %0, %1, off"
                     :: "v"(ly), "v"(gy) : "memory");
    }
}

__global__ __launch_bounds__(256)
void sim_kernel(const float* __restrict__ x, const float* __restrict__ y,
                float* __restrict__ sim) {
    __shared__ float lds[2 * BUFF];      // ping-pong (xs|ys) buffers

    const int t    = threadIdx.x;
    const int lane = t & 31;
    const int wave = t >> 5;
    const int wg   = blockIdx.x;        // 0..1023
    const int b    = wg >> 5;           // batch
    const int cb   = wg & 31;           // 16-channel block
    const size_t base = ((size_t)b * C_ + (size_t)cb * 16) * HW;

    v8f axy = {};
    v8f axx = {};
    v8f ayy = {};

    // WMMA operand addressing: lane L supplies channel (L & 15), two
    // consecutive spatial values starting at s + 2*(L >> 4). Identical layout
    // for the A (16x4, M=channel) and B (4x16, N=channel) f32 operands.
    const int ch      = lane & 15;
    const int shalf   = (lane >> 4) * 2;
    const int waveOff = wave * (TILE / NW);

    // Prologue: fill buffer 0 with tile 0.
    async_fill_tile(x, y, lds, lds + 16 * RS, base, 0, t);

    for (int tile = 0; tile < NT; ++tile) {
        float* cur = lds + (tile & 1) * BUFF;
        if (tile + 1 < NT) {
            float* nxt = lds + ((tile + 1) & 1) * BUFF;
            async_fill_tile(x, y, nxt, nxt + 16 * RS, base, tile + 1, t);
            // 8 newer ops may stay in flight; the older tile's 8 have retired.
            asm volatile("s_wait_asynccnt 0x8" ::: "memory");
        } else {
            asm volatile("s_wait_asynccnt 0x0" ::: "memory");
        }
        __syncthreads();

        const float* xs = cur;
        const float* ys = cur + 16 * RS;
        const int baseOff = ch * RS + waveOff + shalf;
        #pragma unroll
        for (int it = 0; it < (TILE / NW) / 4; ++it) {
            int off = baseOff + it * 4;
            v2f a  = *(const v2f*)(xs + off);   // ds_load_b64
            v2f bb = *(const v2f*)(ys + off);   // ds_load_b64
            // (neg_a, A, neg_b, B, c_mod, C, reuse_a, reuse_b)
            axy = __builtin_amdgcn_wmma_f32_16x16x4_f32(false, a,  false, bb,
                                                        (short)0, axy, false, false);
            axx = __builtin_amdgcn_wmma_f32_16x16x4_f32(false, a,  false, a,
                                                        (short)0, axx, false, false);
            ayy = __builtin_amdgcn_wmma_f32_16x16x4_f32(false, bb, false, bb,
                                                        (short)0, ayy, false, false);
        }
        __syncthreads();   // protects buffer reuse by the next async fill
    }

    // Spill accumulators to LDS: red[wave][acc(3)][vgpr(8)][lane(32)]
    float* red = lds;                    // 8*3*8*32 = 6144 floats, reuse
    #pragma unroll
    for (int r = 0; r < 8; ++r) {
        red[(((wave * 3 + 0) * 8 + r) * 32) + lane] = axy[r];
        red[(((wave * 3 + 1) * 8 + r) * 32) + lane] = axx[r];
        red[(((wave * 3 + 2) * 8 + r) * 32) + lane] = ayy[r];
    }
    __syncthreads();

    // Diagonal (m,m) of the 16x16 f32 C/D layout:
    //   m < 8 : VGPR m,   lane m
    //   m >= 8: VGPR m-8, lane m+16
    if (t < 16) {
        const int m  = t;
        const int lm = (m < 8) ? m : (m + 16);
        const int vg = m & 7;
        float dot = 0.f, sxx = 0.f, syy = 0.f;
        #pragma unroll
        for (int w = 0; w < NW; ++w) {
            dot += red[(((w * 3 + 0) * 8 + vg) * 32) + lm];
            sxx += red[(((w * 3 + 1) * 8 + vg) * 32) + lm];
            syy += red[(((w * 3 + 2) * 8 + vg) * 32) + lm];
        }
        float nx = fmaxf(sqrtf(sxx), EPS);
        float ny = fmaxf(sqrtf(syy), EPS);
        sim[(size_t)b * C_ + (size_t)cb * 16 + m] = dot / (nx * ny);
    }
}

// ---------------- Pass 2: select 256 smallest per batch, softmax, scatter gate
__global__ __launch_bounds__(512)
void gate_kernel(const float* __restrict__ sim, float* __restrict__ gate) {
    __shared__ float sv[C_];
    __shared__ float red[512];
    const int b = blockIdx.x;
    const int c = threadIdx.x;

    float v = sim[b * C_ + c];
    sv[c] = v;
    __syncthreads();

    // rank = #(strictly smaller) + #(equal with lower index)  -> matches
    // jax.lax.top_k(-fea_sim) tie-breaking (lower index wins).
    int rank = 0;
    for (int j = 0; j < C_; ++j) {
        float u = sv[j];
        rank += (u < v) || (u == v && j < c);
    }
    const bool sel = rank < KSEL;

    // max over selected values
    red[c] = sel ? v : -INFINITY;
    __syncthreads();
    for (int s = 256; s > 0; s >>= 1) {
        if (c < s) red[c] = fmaxf(red[c], red[c + s]);
        __syncthreads();
    }
    const float m = red[0];
    __syncthreads();

    // sum of exp over selected
    const float e = sel ? expf(v - m) : 0.f;
    red[c] = e;
    __syncthreads();
    for (int s = 256; s > 0; s >>= 1) {
        if (c < s) red[c] += red[c + s];
        __syncthreads();
    }
    const float denom = red[0];

    gate[b * C_ + c] = e / denom;   // 0 for unselected channels
}

// ---------------- Pass 3: out = gate[b,c] * x  (float4 stream)
__global__ __launch_bounds__(256)
void scale_kernel(const float* __restrict__ x, const float* __restrict__ gate,
                  float* __restrict__ out) {
    const size_t i4 = (size_t)blockIdx.x * 256 + threadIdx.x;  // float4 index
    const size_t bc = i4 >> 10;                                // 1024 f4 per (b,c)
    const float g = gate[bc];
    v4f v = *(const v4f*)(x + i4 * 4);
    *(v4f*)(out + i4 * 4) = v * g;
}

extern "C" void kernel_launch(void* const* d_in, const int* in_sizes, int n_in,
                              void* d_out, int out_size, void* d_ws, size_t ws_size,
                              hipStream_t stream) {
    const float* x = (const float*)d_in[0];
    const float* y = (const float*)d_in[1];
    float* out  = (float*)d_out;
    float* sim  = (float*)d_ws;             // B_*C_ floats
    float* gate = sim + (size_t)B_ * C_;    // B_*C_ floats

    sim_kernel<<<B_ * (C_ / 16), 256, 0, stream>>>(x, y, sim);
    gate_kernel<<<B_, 512, 0, stream>>>(sim, gate);

    const int n4blocks = (int)(((size_t)B_ * C_ * HW / 4) / 256);  // 65536
    scale_kernel<<<n4blocks, 256, 0, stream>>>(x, gate, out);
}